// Patch_QML_Skip_Engine_47167330844885
// MI455X (gfx1250) — compile-verified
//
#include <hip/hip_runtime.h>
#include <hip/hip_bf16.h>

typedef __attribute__((ext_vector_type(2))) float v2f;
typedef __attribute__((ext_vector_type(4))) float v4f;
typedef __attribute__((ext_vector_type(8))) float v8f;

#define SEQ_LEN 512
#define PRED_LEN 96
#define PATCH_LEN 16
#define STRIDE 8
#define NUM_PATCHES 63          // (512-16)/8 + 1
#define BB 32
#define MM 128
#define BM (BB * MM)            // 4096
#define FAN_IN (NUM_PATCHES * 4)  // 252
#define KTOT (FAN_IN + SEQ_LEN)   // 764
#define KPAD 768                  // 24 chunks of 32
#define KC 32
#define KCP 34                    // padded LDS row stride (dwords)

// ---------------------------------------------------------------------------
// Kernel 1: per-patch 4-qubit statevector simulation -> PauliZ expectations.
// One patch per lane; 16 complex amplitudes fully unrolled in registers.
// Rotation matrices (12 of them, from the 36 weight scalars) precomputed
// into LDS once per block.
// ---------------------------------------------------------------------------
__global__ __launch_bounds__(256) void qenc_kernel(const float* __restrict__ x,
                                                   const float* __restrict__ wts,
                                                   float* __restrict__ enc) {
    __shared__ float rotm[12][8];  // [l*4+w][re00,im00,re01,im01,re10,im10,re11,im11]

    const int tid = threadIdx.x;
    if (tid < 12) {
        const int l = tid >> 2, w = tid & 3;
        const float phi = wts[l * 12 + w * 3 + 0];
        const float th  = wts[l * 12 + w * 3 + 1];
        const float om  = wts[l * 12 + w * 3 + 2];
        const float c = __cosf(0.5f * th), s = __sinf(0.5f * th);
        const float ap = 0.5f * (phi + om), am = 0.5f * (phi - om);
        const float ca = __cosf(ap), sa = __sinf(ap);
        const float cb = __cosf(am), sb = __sinf(am);
        rotm[tid][0] =  ca * c;  rotm[tid][1] = -sa * c;   // m00 = e^{-i a} c
        rotm[tid][2] = -cb * s;  rotm[tid][3] = -sb * s;   // m01 = -e^{+i b} s
        rotm[tid][4] =  cb * s;  rotm[tid][5] = -sb * s;   // m10 = e^{-i b} s
        rotm[tid][6] =  ca * c;  rotm[tid][7] =  sa * c;   // m11 = e^{+i a} c
    }
    __syncthreads();

    const int n  = blockIdx.x * 256 + tid;     // 0 .. 258048-1 (grid exact)
    const int p  = n / BM;                     // patch index 0..62
    const int bm = n - p * BM;                 // flat (b,m), consecutive across lanes
    const int b  = bm >> 7;
    const int m  = bm & 127;

    // load patch: x[b, p*8 + j, m], add 1e-6, normalize (amplitude embedding)
    float sr[16], si[16];
    float nrm = 0.0f;
    const size_t base = ((size_t)b * SEQ_LEN + (size_t)p * STRIDE) * MM + m;
#pragma unroll
    for (int j = 0; j < 16; ++j) {
        float v = x[base + (size_t)j * MM] + 1e-6f;
        sr[j] = v;
        nrm += v * v;
    }
    const float inv = rsqrtf(nrm);
#pragma unroll
    for (int j = 0; j < 16; ++j) { sr[j] *= inv; si[j] = 0.0f; }

    // 3 StronglyEntanglingLayers. Qubit w == bit (3-w) of the state index.
#pragma unroll
    for (int l = 0; l < 3; ++l) {
#pragma unroll
        for (int w = 0; w < 4; ++w) {
            const float* U = &rotm[l * 4 + w][0];
            const float u00r = U[0], u00i = U[1], u01r = U[2], u01i = U[3];
            const float u10r = U[4], u10i = U[5], u11r = U[6], u11i = U[7];
            const int mask = 8 >> w;
#pragma unroll
            for (int i = 0; i < 16; ++i) {
                if (i & mask) continue;
                const int i1 = i | mask;
                const float ar = sr[i],  ai = si[i];
                const float br = sr[i1], bi = si[i1];
                sr[i]  = u00r * ar - u00i * ai + u01r * br - u01i * bi;
                si[i]  = u00r * ai + u00i * ar + u01r * bi + u01i * br;
                sr[i1] = u10r * ar - u10i * ai + u11r * br - u11i * bi;
                si[i1] = u10r * ai + u10i * ar + u11r * bi + u11i * br;
            }
        }
        const int rr = (l % 3) + 1;  // CNOT range
#pragma unroll
        for (int w = 0; w < 4; ++w) {
            const int cm = 8 >> w;
            const int tm = 8 >> ((w + rr) & 3);
#pragma unroll
            for (int i = 0; i < 16; ++i) {
                if ((i & cm) && !(i & tm)) {
                    const int i1 = i | tm;
                    float t;
                    t = sr[i]; sr[i] = sr[i1]; sr[i1] = t;
                    t = si[i]; si[i] = si[i1]; si[i1] = t;
                }
            }
        }
    }

    // PauliZ expectation per wire from probabilities
    float z[4] = {0.0f, 0.0f, 0.0f, 0.0f};
#pragma unroll
    for (int i = 0; i < 16; ++i) {
        const float pp = sr[i] * sr[i] + si[i] * si[i];
#pragma unroll
        for (int w = 0; w < 4; ++w)
            z[w] += (i & (8 >> w)) ? -pp : pp;
    }

    v4f zv = {z[0], z[1], z[2], z[3]};
    *(v4f*)&enc[(size_t)bm * FAN_IN + (size_t)p * 4] = zv;  // 16B aligned
}

// ---------------------------------------------------------------------------
// Kernel 2: fused GEMM  out = [enc | x_flat] @ [head_w | skip_w]^T + biases,
// written directly in (B, pred_len, M) order.
// One block per batch b (M-tile = 128 rows of m); 8 waves, each wave does a
// 16(m) x 96(t) tile via 6 f32 WMMA accumulators; K=764 (zero-padded to 768)
// tiled at 32 through LDS.
// ---------------------------------------------------------------------------
__global__ __launch_bounds__(256) void fused_gemm_kernel(
    const float* __restrict__ enc, const float* __restrict__ x,
    const float* __restrict__ head_w, const float* __restrict__ head_b,
    const float* __restrict__ skip_w, const float* __restrict__ skip_b,
    float* __restrict__ out) {
    __shared__ float Alds[128][KCP];  // [m-row][k]
    __shared__ float Blds[96][KCP];   // [t-row][k]

    const int tid = threadIdx.x;
    const int wv   = tid >> 5;         // wave 0..7 -> m rows wv*16..wv*16+15
    const int lane = tid & 31;
    const int h    = lane >> 4;        // K-half select for A/B frags
    const int q    = lane & 15;        // A row / B col within tile

    v8f acc[6];
#pragma unroll
    for (int nn = 0; nn < 6; ++nn) acc[nn] = (v8f)0.0f;

#pragma unroll 1
    for (int k0 = 0; k0 < KPAD; k0 += KC) {
        // stage A: 128 x 32 (row = m, coalesced over m for the x/skip part)
#pragma unroll
        for (int i = 0; i < 16; ++i) {
            const int e   = tid + i * 256;
            const int row = e & 127;
            const int kk  = e >> 7;
            const int k   = k0 + kk;
            float v = 0.0f;
            if (k < FAN_IN)
                v = enc[(size_t)(blockIdx.x * MM + row) * FAN_IN + k];
            else if (k < KTOT)
                v = x[((size_t)blockIdx.x * SEQ_LEN + (k - FAN_IN)) * MM + row];
            Alds[row][kk] = v;
        }
        // stage B: 96 x 32 (row = t, coalesced over k)
#pragma unroll
        for (int i = 0; i < 12; ++i) {
            const int e  = tid + i * 256;
            const int t  = e >> 5;
            const int kk = e & 31;
            const int k  = k0 + kk;
            float v = 0.0f;
            if (k < FAN_IN)
                v = head_w[(size_t)t * FAN_IN + k];
            else if (k < KTOT)
                v = skip_w[(size_t)t * SEQ_LEN + (k - FAN_IN)];
            Blds[t][kk] = v;
        }
        __syncthreads();

#pragma unroll
        for (int kk = 0; kk < KC; kk += 4) {
            // A frag: lane holds row q, K = kk + 2h + {0,1}
            const v2f a = *(const v2f*)&Alds[wv * 16 + q][kk + 2 * h];
#pragma unroll
            for (int nn = 0; nn < 6; ++nn) {
                // B frag: lane holds col q (t = nn*16+q), same K split
                const v2f bf = *(const v2f*)&Blds[nn * 16 + q][kk + 2 * h];
                acc[nn] = __builtin_amdgcn_wmma_f32_16x16x4_f32(
                    /*neg_a=*/false, a, /*neg_b=*/false, bf,
                    /*c_mod=*/(short)0, acc[nn],
                    /*reuse_a=*/false, /*reuse_b=*/false);
            }
        }
        __syncthreads();
    }

    // epilogue: bias + transposed store out[b, t, m]
#pragma unroll
    for (int nn = 0; nn < 6; ++nn) {
        const int t = nn * 16 + q;
        const float bias = head_b[t] + skip_b[t];
        const size_t outbase =
            (size_t)blockIdx.x * (PRED_LEN * MM) + (size_t)t * MM + wv * 16 + 8 * h;
#pragma unroll
        for (int j = 0; j < 8; ++j)        // D: VGPR j -> m-row j + 8h
            out[outbase + j] = acc[nn][j] + bias;
    }
}

extern "C" void kernel_launch(void* const* d_in, const int* in_sizes, int n_in,
                              void* d_out, int out_size, void* d_ws, size_t ws_size,
                              hipStream_t stream) {
    const float* x      = (const float*)d_in[0];  // (32, 512, 128)
    const float* wts    = (const float*)d_in[1];  // (3, 4, 3)
    const float* head_w = (const float*)d_in[2];  // (96, 252)
    const float* head_b = (const float*)d_in[3];  // (96,)
    const float* skip_w = (const float*)d_in[4];  // (96, 512)
    const float* skip_b = (const float*)d_in[5];  // (96,)
    float* out = (float*)d_out;                   // (32, 96, 128)
    float* enc = (float*)d_ws;                    // (4096, 252) scratch

    // 258048 patches / 256 = 1008 blocks (exact)
    qenc_kernel<<<1008, 256, 0, stream>>>(x, wts, enc);
    fused_gemm_kernel<<<BB, 256, 0, stream>>>(enc, x, head_w, head_b,
                                              skip_w, skip_b, out);
}